// DeepONet_49357764166328
// MI455X (gfx1250) — compile-verified
//
#include <hip/hip_runtime.h>
#include <hip/hip_bf16.h>
#include <math.h>

#define N_NODES 50000
#define N_EDGES 800000
#define N_GRAPHS 64
#define TPB 256

typedef __attribute__((ext_vector_type(2))) float v2f;
typedef __attribute__((ext_vector_type(8))) float v8f;

static inline int cdiv(long long a, int b) { return (int)((a + b - 1) / b); }

// ---------------- utility ----------------
__global__ void fill_f32(float* __restrict__ p, float v, int n) {
    int i = blockIdx.x * blockDim.x + threadIdx.x;
    if (i < n) p[i] = v;
}

// xl[n,128] = x[n,2] @ W1[2,128]
__global__ void lin1_kernel(const float* __restrict__ x, const float* __restrict__ W1,
                            float* __restrict__ xl, int n) {
    int i = blockIdx.x * blockDim.x + threadIdx.x;
    if (i >= n * 128) return;
    int node = i >> 7, f = i & 127;
    xl[i] = x[node * 2] * W1[f] + x[node * 2 + 1] * W1[128 + f];
}

// es[n,H] = sum_c xl[n,h,c]*as[h,c]; ed likewise
__global__ void escore_kernel(const float* __restrict__ xl, const float* __restrict__ as,
                              const float* __restrict__ ad, float* __restrict__ es,
                              float* __restrict__ ed, int n, int H, int C) {
    int i = blockIdx.x * blockDim.x + threadIdx.x;
    if (i >= n * H) return;
    int node = i / H, h = i % H;
    const float* row = xl + (long long)node * H * C + h * C;
    float s = 0.f, d = 0.f;
    for (int c = 0; c < C; ++c) { s += row[c] * as[h * C + c]; d += row[c] * ad[h * C + c]; }
    es[i] = s; ed[i] = d;
}

__device__ __forceinline__ void atomicMaxF(float* addr, float val) {
    if (val >= 0.f) atomicMax((int*)addr, __float_as_int(val));
    else            atomicMin((unsigned int*)addr, (unsigned int)__float_as_int(val));
}

// pass 1: per-dst per-head max of leaky-relu logits (incl. self loops)
__global__ void edge_max_kernel(const int* __restrict__ src, const int* __restrict__ dst,
                                const float* __restrict__ es, const float* __restrict__ ed,
                                float* __restrict__ emax, int E, int n, int H) {
    int i = blockIdx.x * blockDim.x + threadIdx.x;
    if (i >= (E + n) * H) return;
    int e = i / H, h = i % H;
    int s = e < E ? src[e] : e - E;
    int d = e < E ? dst[e] : e - E;
    float v = es[s * H + h] + ed[d * H + h];
    v = v > 0.f ? v : 0.2f * v;
    atomicMaxF(&emax[d * H + h], v);
}

// pass 2: accumulate exp-weighted features (numerator) and denominators
// one thread per (edge, 4-feature chunk); F = H*C features per edge
__global__ void edge_accum_kernel(const int* __restrict__ src, const int* __restrict__ dst,
                                  const float* __restrict__ es, const float* __restrict__ ed,
                                  const float* __restrict__ emax, const float* __restrict__ xl,
                                  float* __restrict__ denom, float* __restrict__ acc,
                                  int E, int n, int H, int C) {
    int F = H * C;
    int tpe = F >> 2;
    long long i = (long long)blockIdx.x * blockDim.x + threadIdx.x;
    long long total = (long long)(E + n) * tpe;
    if (i >= total) return;
    int e = (int)(i / tpe);
    int f0 = ((int)(i % tpe)) * 4;
    int h = f0 / C;
    int s = e < E ? src[e] : e - E;
    int d = e < E ? dst[e] : e - E;
    float v = es[s * H + h] + ed[d * H + h];
    v = v > 0.f ? v : 0.2f * v;
    float ex = expf(v - emax[d * H + h]);
    if ((f0 % C) == 0) atomicAdd(&denom[d * H + h], ex);
    const float4 xv = *(const float4*)(xl + (long long)s * F + f0);
    float* ap = acc + (long long)d * F + f0;
    atomicAdd(ap + 0, ex * xv.x);
    atomicAdd(ap + 1, ex * xv.y);
    atomicAdd(ap + 2, ex * xv.z);
    atomicAdd(ap + 3, ex * xv.w);
}

// acc <- ELU(acc/denom + bias) in place
__global__ void finalize_elu_kernel(float* __restrict__ acc, const float* __restrict__ denom,
                                    const float* __restrict__ bias, int n, int H, int C) {
    int i = blockIdx.x * blockDim.x + threadIdx.x;
    int F = H * C;
    if (i >= n * F) return;
    int node = i / F, f = i % F, h = f / C;
    float v = acc[i] / fmaxf(denom[node * H + h], 1e-16f) + bias[f];
    acc[i] = v > 0.f ? v : (expf(v) - 1.f);
}

// D[M,Ncols] = A[M,K] @ Bm[K,Ncols], fp32 WMMA 16x16x4. M%16==0, K%4==0, Ncols%16==0.
__global__ void gemm_wmma_f32(const float* __restrict__ A, const float* __restrict__ Bm,
                              float* __restrict__ D, int M, int K, int Ncols) {
    int wave = blockIdx.x * (blockDim.x >> 5) + (threadIdx.x >> 5);
    int ntn = Ncols >> 4;
    int mtile = wave / ntn;
    int ntile = wave % ntn;
    if (mtile * 16 >= M) return;              // wave-uniform: EXEC stays all-1s
    int lane = threadIdx.x & 31;
    int l15 = lane & 15;
    int khalf = (lane >> 4) << 1;             // 0 or 2
    int mrow = mtile * 16 + l15;
    int ncol = ntile * 16 + l15;
    v8f c = {};
    for (int k0 = 0; k0 < K; k0 += 4) {
        v2f a, b;
        a.x = A[(long long)mrow * K + k0 + khalf];
        a.y = A[(long long)mrow * K + k0 + khalf + 1];
        b.x = Bm[(k0 + khalf) * Ncols + ncol];
        b.y = Bm[(k0 + khalf + 1) * Ncols + ncol];
        c = __builtin_amdgcn_wmma_f32_16x16x4_f32(false, a, false, b, (short)0, c,
                                                  false, false);
    }
    int mbase = mtile * 16 + ((lane >> 4) << 3);
    for (int v = 0; v < 8; ++v)
        D[(long long)(mbase + v) * Ncols + ncol] = c[v];
}

// mean pool by graph id
__global__ void pool_kernel(const float* __restrict__ h, const int* __restrict__ batch,
                            float* __restrict__ pooled, float* __restrict__ cnt, int n, int C) {
    int i = blockIdx.x * blockDim.x + threadIdx.x;
    if (i >= n * C) return;
    int node = i / C, c = i % C;
    int b = batch[node];
    atomicAdd(&pooled[b * C + c], h[i]);
    if (c == 0) atomicAdd(&cnt[b], 1.f);
}

// per-graph 64->64 relu -> 64 MLP; grid = B blocks of 64 threads
__global__ void mlp_kernel(const float* __restrict__ pooled, const float* __restrict__ cnt,
                           const float* __restrict__ M1w, const float* __restrict__ M1b,
                           const float* __restrict__ M2w, const float* __restrict__ M2b,
                           float* __restrict__ out) {
    __shared__ float row[64], hid[64];
    int b = blockIdx.x, j = threadIdx.x;
    row[j] = pooled[b * 64 + j] / fmaxf(cnt[b], 1.f);
    __syncthreads();
    float s = M1b[j];
    for (int k = 0; k < 64; ++k) s += row[k] * M1w[k * 64 + j];
    hid[j] = fmaxf(s, 0.f);
    __syncthreads();
    s = M2b[j];
    for (int k = 0; k < 64; ++k) s += hid[k] * M2w[k * 64 + j];
    out[b * 64 + j] = s;
}

// combined = branch*trunk; relu(@Fw1+Fb1); tanh(@Fw2+Fb2) -> [B,2]
__global__ void final_kernel(const float* __restrict__ bout, const float* __restrict__ tout,
                             const float* __restrict__ Fw1, const float* __restrict__ Fb1,
                             const float* __restrict__ Fw2, const float* __restrict__ Fb2,
                             float* __restrict__ out) {
    __shared__ float comb[64], hid[64];
    int b = blockIdx.x, j = threadIdx.x;
    comb[j] = bout[b * 64 + j] * tout[b * 64 + j];
    __syncthreads();
    float s = Fb1[j];
    for (int k = 0; k < 64; ++k) s += comb[k] * Fw1[k * 64 + j];
    hid[j] = fmaxf(s, 0.f);
    __syncthreads();
    if (j < 2) {
        s = Fb2[j];
        for (int k = 0; k < 64; ++k) s += hid[k] * Fw2[k * 2 + j];
        out[b * 2 + j] = tanhf(s);
    }
}

extern "C" void kernel_launch(void* const* d_in, const int* in_sizes, int n_in,
                              void* d_out, int out_size, void* d_ws, size_t ws_size,
                              hipStream_t stream) {
    const int N = N_NODES, E = N_EDGES;

    // -------- workspace partition (floats) --------
    float* ws = (float*)d_ws;
    size_t o = 0;
    float* xlA   = ws + o; o += (size_t)N * 128;  // layer1 xl, reused as layer2 h2lin
    float* acc1  = ws + o; o += (size_t)N * 128;  // layer1 accumulator -> h1
    float* acc2  = ws + o; o += (size_t)N * 64;   // layer2 accumulator -> h2
    float* es1   = ws + o; o += (size_t)N * 2;
    float* ed1   = ws + o; o += (size_t)N * 2;
    float* emax1 = ws + o; o += (size_t)N * 2;
    float* den1  = ws + o; o += (size_t)N * 2;
    float* es2   = ws + o; o += (size_t)N;
    float* ed2   = ws + o; o += (size_t)N;
    float* emax2 = ws + o; o += (size_t)N;
    float* den2  = ws + o; o += (size_t)N;
    float* pooled = ws + o; o += 64 * 64;
    float* cnt    = ws + o; o += 64;
    float* boutp  = ws + o; o += 64 * 64;
    float* toutp  = ws + o; o += 64 * 64;

    auto run_net = [&](int bi, float* netout) {
        const float* x    = (const float*)d_in[bi + 0];
        const int*   ei   = (const int*)  d_in[bi + 1];
        const int*   srcp = ei;
        const int*   dstp = ei + E;
        const int*   batch= (const int*)  d_in[bi + 2];
        const float* W1   = (const float*)d_in[bi + 3];
        const float* a1s  = (const float*)d_in[bi + 4];
        const float* a1d  = (const float*)d_in[bi + 5];
        const float* b1   = (const float*)d_in[bi + 6];
        const float* W2   = (const float*)d_in[bi + 7];
        const float* a2s  = (const float*)d_in[bi + 8];
        const float* a2d  = (const float*)d_in[bi + 9];
        const float* b2   = (const float*)d_in[bi + 10];
        const float* M1w  = (const float*)d_in[bi + 11];
        const float* M1b  = (const float*)d_in[bi + 12];
        const float* M2w  = (const float*)d_in[bi + 13];
        const float* M2b  = (const float*)d_in[bi + 14];

        const float NEG_INF = -__builtin_huge_valf();

        // init accumulators
        fill_f32<<<cdiv((long long)N * 128, TPB), TPB, 0, stream>>>(acc1, 0.f, N * 128);
        fill_f32<<<cdiv(N * 2, TPB), TPB, 0, stream>>>(den1, 0.f, N * 2);
        fill_f32<<<cdiv(N * 2, TPB), TPB, 0, stream>>>(emax1, NEG_INF, N * 2);
        fill_f32<<<cdiv((long long)N * 64, TPB), TPB, 0, stream>>>(acc2, 0.f, N * 64);
        fill_f32<<<cdiv(N, TPB), TPB, 0, stream>>>(den2, 0.f, N);
        fill_f32<<<cdiv(N, TPB), TPB, 0, stream>>>(emax2, NEG_INF, N);
        fill_f32<<<cdiv(64 * 64, TPB), TPB, 0, stream>>>(pooled, 0.f, 64 * 64);
        fill_f32<<<1, 64, 0, stream>>>(cnt, 0.f, 64);

        // ---- GAT layer 1 (H=2, C=64) ----
        lin1_kernel<<<cdiv((long long)N * 128, TPB), TPB, 0, stream>>>(x, W1, xlA, N);
        escore_kernel<<<cdiv(N * 2, TPB), TPB, 0, stream>>>(xlA, a1s, a1d, es1, ed1, N, 2, 64);
        edge_max_kernel<<<cdiv((long long)(E + N) * 2, TPB), TPB, 0, stream>>>(
            srcp, dstp, es1, ed1, emax1, E, N, 2);
        edge_accum_kernel<<<cdiv((long long)(E + N) * 32, TPB), TPB, 0, stream>>>(
            srcp, dstp, es1, ed1, emax1, xlA, den1, acc1, E, N, 2, 64);
        finalize_elu_kernel<<<cdiv((long long)N * 128, TPB), TPB, 0, stream>>>(
            acc1, den1, b1, N, 2, 64);

        // ---- GAT layer 2 (H=1, C=64): h2lin = h1 @ W2 via fp32 WMMA ----
        {
            int tiles = (N / 16) * (64 / 16);          // 3125 * 4 = 12500
            int waves_per_block = TPB / 32;
            gemm_wmma_f32<<<cdiv(tiles, waves_per_block), TPB, 0, stream>>>(
                acc1, W2, xlA, N, 128, 64);
        }
        escore_kernel<<<cdiv(N, TPB), TPB, 0, stream>>>(xlA, a2s, a2d, es2, ed2, N, 1, 64);
        edge_max_kernel<<<cdiv((long long)(E + N), TPB), TPB, 0, stream>>>(
            srcp, dstp, es2, ed2, emax2, E, N, 1);
        edge_accum_kernel<<<cdiv((long long)(E + N) * 16, TPB), TPB, 0, stream>>>(
            srcp, dstp, es2, ed2, emax2, xlA, den2, acc2, E, N, 1, 64);
        finalize_elu_kernel<<<cdiv((long long)N * 64, TPB), TPB, 0, stream>>>(
            acc2, den2, b2, N, 1, 64);

        // ---- pooling + per-graph MLP ----
        pool_kernel<<<cdiv((long long)N * 64, TPB), TPB, 0, stream>>>(
            acc2, batch, pooled, cnt, N, 64);
        mlp_kernel<<<N_GRAPHS, 64, 0, stream>>>(pooled, cnt, M1w, M1b, M2w, M2b, netout);
    };

    run_net(0, boutp);    // branch inputs: d_in[0..14]
    run_net(15, toutp);   // trunk inputs:  d_in[15..29]

    const float* Fw1 = (const float*)d_in[30];
    const float* Fb1 = (const float*)d_in[31];
    const float* Fw2 = (const float*)d_in[32];
    const float* Fb2 = (const float*)d_in[33];
    final_kernel<<<N_GRAPHS, 64, 0, stream>>>(boutp, toutp, Fw1, Fb1, Fw2, Fb2, (float*)d_out);
}